// _NNConv_54408645705759
// MI455X (gfx1250) — compile-verified
//
#include <hip/hip_runtime.h>

// NNConv (edge-conditioned conv) — algebraically collapsed.
// Since the reference only consumes m via m.sum(axis=1) (scalar mean broadcast),
// the E x 64 x 64 per-edge weight tensor never needs to exist:
//   msum[e] = sum_d ea[e,d] * (Wsum[d] . x[row[e]]) + bsum . x[row[e]]
// with Wsum[d,c] = sum_o W_nn[d, c*64+o], bsum[c] = sum_o b_nn[c*64+o].
// Remaining dense GEMM (out = x @ root) runs on V_WMMA_F32_16X16X4_F32.
//
// Workspace layout (floats):
//   acc  [0      , 10000)   per-node segment sum of msum
//   cnt  [10000  , 20000)   per-node incoming-edge count
//   Wsum [20000  , 21024)   16 x 64
//   bsum [21024  , 21088)   64
//   v    [21088  , 191088)  10000 x 17 node projections
// Total ~765 KB.

#define NN  10000
#define NE  100000
#define CC  64
#define DE  16

typedef __attribute__((ext_vector_type(2))) float v2f;
typedef __attribute__((ext_vector_type(8))) float v8f;

__global__ void _nnconv_zero(float* __restrict__ p, int n) {
    int i = blockIdx.x * blockDim.x + threadIdx.x;
    if (i < n) p[i] = 0.0f;
}

// Wsum[d,c] = sum_o W_nn[d, c*64+o]; bsum[c] = sum_o b_nn[c*64+o]
__global__ void _nnconv_wsum(const float* __restrict__ Wnn,
                             const float* __restrict__ bnn,
                             float* __restrict__ Wsum,
                             float* __restrict__ bsum) {
    int t = blockIdx.x * blockDim.x + threadIdx.x;
    if (t < DE * CC) {
        int d = t >> 6, c = t & 63;
        const float* p = Wnn + d * (CC * CC) + c * CC;
        float s = 0.0f;
        for (int o = 0; o < CC; ++o) s += p[o];
        Wsum[t] = s;
    } else if (t < DE * CC + CC) {
        int c = t - DE * CC;
        const float* p = bnn + c * CC;
        float s = 0.0f;
        for (int o = 0; o < CC; ++o) s += p[o];
        bsum[c] = s;
    }
}

// v[n, d<16] = Wsum[d] . x[n];  v[n, 16] = bsum . x[n]
__global__ void _nnconv_nodev(const float* __restrict__ x,
                              const float* __restrict__ Wsum,
                              const float* __restrict__ bsum,
                              float* __restrict__ v) {
    int t = blockIdx.x * blockDim.x + threadIdx.x;
    if (t >= NN * 17) return;
    int n = t / 17;
    int d = t - n * 17;
    const float* xr = x + n * CC;
    const float* w  = (d < 16) ? (Wsum + d * CC) : bsum;
    float s = 0.0f;
    for (int c = 0; c < CC; ++c) s += xr[c] * w[c];
    v[t] = s;
}

// msum[e] = v[row,16] + sum_d ea[e,d]*v[row,d]; atomically segment-reduce by col.
__global__ void _nnconv_edge(const int* __restrict__ ei,
                             const float* __restrict__ ea,
                             const float* __restrict__ v,
                             float* __restrict__ acc,
                             float* __restrict__ cnt) {
    int e = blockIdx.x * blockDim.x + threadIdx.x;
    if (e >= NE) return;
    int r = ei[e];
    int c = ei[NE + e];
    const float* vr = v + r * 17;
    const float* a  = ea + e * DE;
    float s = vr[16];
#pragma unroll
    for (int d = 0; d < DE; ++d) s += a[d] * vr[d];
    unsafeAtomicAdd(&acc[c], s);
    unsafeAtomicAdd(&cnt[c], 1.0f);
}

// out[n,:] = x[n,:] @ root + bias + mean[n], via V_WMMA_F32_16X16X4_F32.
// One wave per 16-node tile; 4 output tiles of 16 cols; K=64 in steps of 4.
__global__ void _nnconv_out_wmma(const float* __restrict__ x,
                                 const float* __restrict__ root,
                                 const float* __restrict__ bias,
                                 const float* __restrict__ acc,
                                 const float* __restrict__ cnt,
                                 float* __restrict__ out) {
    const int wave = blockIdx.x * (blockDim.x >> 5) + (threadIdx.x >> 5);
    if (wave >= NN / 16) return;            // wave-uniform: EXEC stays all-ones
    const int lane = threadIdx.x & 31;
    const int half = lane >> 4;             // 0: lanes 0-15, 1: lanes 16-31
    const int l16  = lane & 15;
    const int nb   = wave * 16;             // node base (16 | 10000 exactly)

    v8f d0 = {}, d1 = {}, d2 = {}, d3 = {};
    const float* xrow = x + (nb + l16) * CC;

#pragma unroll
    for (int k0 = 0; k0 < CC; k0 += 4) {
        // fp32 A 16x4 layout: v0 = K=(2*half), v1 = K=(2*half+1), M = l16
        const int ka = k0 + 2 * half;
        v2f a;
        a.x = xrow[ka];
        a.y = xrow[ka + 1];
        const float* r0 = root + ka * CC;
        const float* r1 = root + (ka + 1) * CC;
        v2f b;
        b.x = r0[l16];       b.y = r1[l16];
        d0 = __builtin_amdgcn_wmma_f32_16x16x4_f32(false, a, false, b, (short)0, d0, false, false);
        b.x = r0[16 + l16];  b.y = r1[16 + l16];
        d1 = __builtin_amdgcn_wmma_f32_16x16x4_f32(false, a, false, b, (short)0, d1, false, false);
        b.x = r0[32 + l16];  b.y = r1[32 + l16];
        d2 = __builtin_amdgcn_wmma_f32_16x16x4_f32(false, a, false, b, (short)0, d2, false, false);
        b.x = r0[48 + l16];  b.y = r1[48 + l16];
        d3 = __builtin_amdgcn_wmma_f32_16x16x4_f32(false, a, false, b, (short)0, d3, false, false);
    }

    const float b0 = bias[l16];
    const float b1 = bias[16 + l16];
    const float b2 = bias[32 + l16];
    const float b3 = bias[48 + l16];

#pragma unroll
    for (int g = 0; g < 8; ++g) {
        // C/D layout: VGPR g holds row M=g (lanes 0-15) / M=g+8 (lanes 16-31)
        const int node = nb + g + 8 * half;
        const float cv = cnt[node];
        const float mv = (cv > 0.5f) ? (acc[node] / (cv * (float)CC)) : 0.0f;
        float* orow = out + node * CC;
        orow[l16]      = d0[g] + b0 + mv;
        orow[16 + l16] = d1[g] + b1 + mv;
        orow[32 + l16] = d2[g] + b2 + mv;
        orow[48 + l16] = d3[g] + b3 + mv;
    }
}

extern "C" void kernel_launch(void* const* d_in, const int* in_sizes, int n_in,
                              void* d_out, int out_size, void* d_ws, size_t ws_size,
                              hipStream_t stream) {
    const float* x    = (const float*)d_in[0];   // [10000, 64]
    const int*   ei   = (const int*)  d_in[1];   // [2, 100000]
    const float* ea   = (const float*)d_in[2];   // [100000, 16]
    const float* Wnn  = (const float*)d_in[3];   // [16, 4096]
    const float* bnn  = (const float*)d_in[4];   // [4096]
    const float* root = (const float*)d_in[5];   // [64, 64]
    const float* bias = (const float*)d_in[6];   // [64]
    float* out = (float*)d_out;                  // [10000, 64]

    float* ws   = (float*)d_ws;
    float* acc  = ws;
    float* cnt  = ws + NN;
    float* Wsum = ws + 2 * NN;
    float* bsum = Wsum + DE * CC;
    float* v    = bsum + CC;

    // 1. zero the segment accumulators (deterministic across graph replays)
    _nnconv_zero<<<(2 * NN + 255) / 256, 256, 0, stream>>>(acc, 2 * NN);
    // 2. reduce W_nn / b_nn over the output axis
    _nnconv_wsum<<<(DE * CC + CC + 255) / 256, 256, 0, stream>>>(Wnn, bnn, Wsum, bsum);
    // 3. per-node 17-way projection
    _nnconv_nodev<<<(NN * 17 + 255) / 256, 256, 0, stream>>>(x, Wsum, bsum, v);
    // 4. per-edge scalar message + atomic segment sum
    _nnconv_edge<<<(NE + 255) / 256, 256, 0, stream>>>(ei, ea, v, acc, cnt);
    // 5. WMMA root transform + bias + broadcast mean (625 waves, 4 waves/block)
    _nnconv_out_wmma<<<(NN / 16 + 3) / 4, 128, 0, stream>>>(x, root, bias, acc, cnt, out);
}